// QuantumKernelLayer_38620345926120
// MI455X (gfx1250) — compile-verified
//
#include <hip/hip_runtime.h>
#include <hip/hip_bf16.h>
#include <math.h>

typedef float v2f __attribute__((ext_vector_type(2)));
typedef float v8f __attribute__((ext_vector_type(8)));

#define INV_SQRT2F 0.70710678118654752440f

// ---------------------------------------------------------------------------
// Forward feature-map circuit W(x) on a 64-amplitude state held as 2 complex
// amplitudes per lane of a wave32 (slot0 = index lane, slot1 = index lane+32).
// Qubit q acts on bit (5-q) of the state index.
// ---------------------------------------------------------------------------
__device__ __forceinline__ void apply_circuit(float& ar0, float& ai0,
                                              float& ar1, float& ai1,
                                              const float* __restrict__ x,
                                              const float* __restrict__ params,
                                              int lane) {
  for (int rep = 0; rep < 2; ++rep) {
    for (int q = 0; q < 6; ++q) {
      const int b = 5 - q;
      // ---- Hadamard ----
      if (b == 5) {
        float nr0 = (ar0 + ar1) * INV_SQRT2F, ni0 = (ai0 + ai1) * INV_SQRT2F;
        float nr1 = (ar0 - ar1) * INV_SQRT2F, ni1 = (ai0 - ai1) * INV_SQRT2F;
        ar0 = nr0; ai0 = ni0; ar1 = nr1; ai1 = ni1;
      } else {
        const int m = 1 << b;
        float pr0 = __shfl_xor(ar0, m, 32), pi0 = __shfl_xor(ai0, m, 32);
        float pr1 = __shfl_xor(ar1, m, 32), pi1 = __shfl_xor(ai1, m, 32);
        if (lane & m) {
          ar0 = (pr0 - ar0) * INV_SQRT2F; ai0 = (pi0 - ai0) * INV_SQRT2F;
          ar1 = (pr1 - ar1) * INV_SQRT2F; ai1 = (pi1 - ai1) * INV_SQRT2F;
        } else {
          ar0 = (ar0 + pr0) * INV_SQRT2F; ai0 = (ai0 + pi0) * INV_SQRT2F;
          ar1 = (ar1 + pr1) * INV_SQRT2F; ai1 = (ai1 + pi1) * INV_SQRT2F;
        }
      }
      // ---- RZ: bit0 -> e^{-i t/2}, bit1 -> e^{+i t/2} ----
      const float half_t = 0.5f * (x[q] + params[rep * 6 + q]);
      float sn, cs;
      sincosf(half_t, &sn, &cs);
      float sg0, sg1;
      if (b == 5) { sg0 = -1.0f; sg1 = 1.0f; }
      else        { float sg = ((lane >> b) & 1) ? 1.0f : -1.0f; sg0 = sg; sg1 = sg; }
      float t;
      t   = ar0 * cs - ai0 * (sg0 * sn);
      ai0 = ar0 * (sg0 * sn) + ai0 * cs; ar0 = t;
      t   = ar1 * cs - ai1 * (sg1 * sn);
      ai1 = ar1 * (sg1 * sn) + ai1 * cs; ar1 = t;
    }
    // ---- CNOT chain: control bit (5-c), target bit (4-c) ----
    for (int cq = 0; cq < 5; ++cq) {
      const int cb = 5 - cq, tb = 4 - cq;
      if (cb == 5) {
        ar1 = __shfl_xor(ar1, 1 << tb, 32);
        ai1 = __shfl_xor(ai1, 1 << tb, 32);
      } else {
        const int tm = 1 << tb;
        float pr0 = __shfl_xor(ar0, tm, 32), pi0 = __shfl_xor(ai0, tm, 32);
        float pr1 = __shfl_xor(ar1, tm, 32), pi1 = __shfl_xor(ai1, tm, 32);
        if ((lane >> cb) & 1) { ar0 = pr0; ai0 = pi0; ar1 = pr1; ai1 = pi1; }
      }
    }
  }
}

// ---------------------------------------------------------------------------
// CDNA5 async global->LDS copy of 16 bytes (tracked by ASYNCcnt).
// GVS addressing: mem = SGPR64 + VGPR32 byte offset. LDS dest offset is the
// low 32 bits of the generic pointer to the __shared__ element.
// ---------------------------------------------------------------------------
__device__ __forceinline__ void async_b128(const float* __restrict__ gbase,
                                           unsigned gbyte_off,
                                           const float* lds_ptr) {
  unsigned lds_off = (unsigned)(unsigned long long)(const void*)lds_ptr;
  asm volatile("global_load_async_to_lds_b128 %0, %1, %2"
               :
               : "v"(lds_off), "v"(gbyte_off), "s"(gbase)
               : "memory");
}

__device__ __forceinline__ void wait_async0() {
  asm volatile("s_wait_asynccnt 0" ::: "memory");
}

// ---------------------------------------------------------------------------
// Phase 1: psi_i = W(x_i)|0> for all 256 samples, stored [i=256][k=64].
// ---------------------------------------------------------------------------
__global__ void k_states(const float* __restrict__ X,
                         const float* __restrict__ params,
                         float* __restrict__ PsiR,
                         float* __restrict__ PsiI) {
  const int wid  = (blockIdx.x * blockDim.x + threadIdx.x) >> 5; // sample i
  const int lane = threadIdx.x & 31;
  float ar0 = (lane == 0) ? 1.0f : 0.0f, ai0 = 0.0f;
  float ar1 = 0.0f, ai1 = 0.0f;
  apply_circuit(ar0, ai0, ar1, ai1, X + wid * 6, params, lane);
  PsiR[wid * 64 + lane]      = ar0;
  PsiR[wid * 64 + 32 + lane] = ar1;
  PsiI[wid * 64 + lane]      = ai0;
  PsiI[wid * 64 + 32 + lane] = ai1;
}

// ---------------------------------------------------------------------------
// Phase 2: A_j = W(x_j)^dagger stored [j][s=64][k=64]:
//   Ar = Re(W[k,s]), Aim = -Im(W[k,s]) (conjugate folded in).
// ---------------------------------------------------------------------------
__global__ void k_unitary(const float* __restrict__ X,
                          const float* __restrict__ params,
                          float* __restrict__ Ar,
                          float* __restrict__ Aim) {
  const int gw   = (blockIdx.x * blockDim.x + threadIdx.x) >> 5;
  const int lane = threadIdx.x & 31;
  const int j = gw >> 6;
  const int s = gw & 63;
  float ar0 = (s == lane)      ? 1.0f : 0.0f, ai0 = 0.0f;
  float ar1 = (s == lane + 32) ? 1.0f : 0.0f, ai1 = 0.0f;
  apply_circuit(ar0, ai0, ar1, ai1, X + j * 6, params, lane);
  float* arow = Ar  + ((size_t)j * 64 + s) * 64;
  float* irow = Aim + ((size_t)j * 64 + s) * 64;
  arow[lane]      = ar0;
  arow[32 + lane] = ar1;
  irow[lane]      = -ai0;
  irow[32 + lane] = -ai1;
}

// ---------------------------------------------------------------------------
// Phase 3: per j, complex GEMM V = A_j * Psi with V_WMMA_F32_16X16X4_F32,
// operands staged into LDS via async b128 copies:
//   Vr = Ar*PsiR + Aim*(-PsiI),  Vi = Ar*PsiI + Aim*PsiR
// LDS rows padded to 68 floats: 272B row pitch keeps b128 writes 16B-aligned
// and makes fragment reads bank-conflict-free (bank = (4*row + k) mod 64).
// ---------------------------------------------------------------------------
#define PITCH 68

__global__ void __launch_bounds__(256)
k_gemm(const float* __restrict__ Ar,
       const float* __restrict__ Aim,
       const float* __restrict__ PsiR,
       const float* __restrict__ PsiI,
       float* __restrict__ Kraw) {
  __shared__ float sAr[64 * PITCH];    //  17 KB
  __shared__ float sAi[64 * PITCH];    //  17 KB
  __shared__ float sPr[256 * PITCH];   //  69.6 KB
  __shared__ float sPi[256 * PITCH];   //  69.6 KB

  const int j    = blockIdx.x;
  const int t    = threadIdx.x;
  const int wave = t >> 5;
  const int lane = t & 31;
  const int half = lane >> 4;
  const int l16  = lane & 15;

  // ---- async stage: Psi (4096 16B chunks per matrix), A_j (1024 per) ----
  const float* Ajg = Ar  + (size_t)j * 64 * 64;
  const float* Amg = Aim + (size_t)j * 64 * 64;
  for (int c = t; c < 4096; c += 256) {
    const int row = c >> 4, k4 = (c & 15) << 2;
    async_b128(PsiR, (unsigned)c * 16u, &sPr[row * PITCH + k4]);
    async_b128(PsiI, (unsigned)c * 16u, &sPi[row * PITCH + k4]);
  }
  for (int c = t; c < 1024; c += 256) {
    const int row = c >> 4, k4 = (c & 15) << 2;
    async_b128(Ajg, (unsigned)c * 16u, &sAr[row * PITCH + k4]);
    async_b128(Amg, (unsigned)c * 16u, &sAi[row * PITCH + k4]);
  }
  wait_async0();
  __syncthreads();

  // ---- WMMA complex GEMM + measurement contraction ----
  for (int icc = 0; icc < 2; ++icc) {
    const int ic  = wave * 2 + icc;
    const int col = ic * 16 + l16;     // sample index i for this lane
    float meas[5] = {0.f, 0.f, 0.f, 0.f, 0.f};

    for (int sr = 0; sr < 4; ++sr) {
      const int sbase = sr * 16;
      v8f vr = {0.f, 0.f, 0.f, 0.f, 0.f, 0.f, 0.f, 0.f};
      v8f vi = {0.f, 0.f, 0.f, 0.f, 0.f, 0.f, 0.f, 0.f};
      const int arow = sbase + l16;

      #pragma unroll 4
      for (int kk = 0; kk < 16; ++kk) {
        const int ks = kk * 4 + 2 * half;   // per-lane K offsets
        v2f a  = *(const v2f*)(sAr + arow * PITCH + ks);
        v2f am = *(const v2f*)(sAi + arow * PITCH + ks);
        v2f br = *(const v2f*)(sPr + col  * PITCH + ks);
        v2f bi = *(const v2f*)(sPi + col  * PITCH + ks);
        v2f bn = { -bi.x, -bi.y };
        vr = __builtin_amdgcn_wmma_f32_16x16x4_f32(false, a,  false, br, (short)0, vr, false, false);
        vr = __builtin_amdgcn_wmma_f32_16x16x4_f32(false, am, false, bn, (short)0, vr, false, false);
        vi = __builtin_amdgcn_wmma_f32_16x16x4_f32(false, a,  false, bi, (short)0, vi, false, false);
        vi = __builtin_amdgcn_wmma_f32_16x16x4_f32(false, am, false, br, (short)0, vi, false, false);
      }

      // C/D layout: VGPR r, this lane -> state s = sbase + r + 8*half
      #pragma unroll
      for (int r = 0; r < 8; ++r) {
        const float p = vr[r] * vr[r] + vi[r] * vi[r];
        const int s = sbase + r + 8 * half;
        #pragma unroll
        for (int pp = 0; pp < 5; ++pp) {
          const int neg = ((s >> (5 - pp)) ^ (s >> (4 - pp))) & 1;
          meas[pp] += neg ? -p : p;
        }
      }
    }

    float ksum = 0.f;
    #pragma unroll
    for (int pp = 0; pp < 5; ++pp) {
      const float m = meas[pp] + __shfl_xor(meas[pp], 16, 32);
      ksum += fabsf(m);
    }
    ksum *= 0.2f;
    if (half == 0) Kraw[(size_t)col * 256 + j] = ksum;  // row = x1 sample i
  }
}

// ---------------------------------------------------------------------------
// Phase 4: K[i][j] = (i <= j) ? Kraw[i][j] : Kraw[j][i]
// ---------------------------------------------------------------------------
__global__ void k_symmetrize(const float* __restrict__ Kraw,
                             float* __restrict__ Out) {
  const int i  = blockIdx.x;
  const int jj = threadIdx.x;
  Out[(size_t)i * 256 + jj] =
      (i <= jj) ? Kraw[(size_t)i * 256 + jj] : Kraw[(size_t)jj * 256 + i];
}

extern "C" void kernel_launch(void* const* d_in, const int* in_sizes, int n_in,
                              void* d_out, int out_size, void* d_ws, size_t ws_size,
                              hipStream_t stream) {
  (void)in_sizes; (void)n_in; (void)out_size; (void)ws_size;
  const float* X      = (const float*)d_in[0];   // 256 x 6
  const float* params = (const float*)d_in[1];   // 2 x 6
  float* out          = (float*)d_out;           // 256 x 256

  // workspace carve-up (floats)
  float* ws   = (float*)d_ws;
  float* PsiR = ws;                       // 256*64
  float* PsiI = PsiR + 256 * 64;          // 256*64
  float* Ar   = PsiI + 256 * 64;          // 256*64*64
  float* Aim  = Ar + 256 * 64 * 64;       // 256*64*64
  float* Kraw = Aim + 256 * 64 * 64;      // 256*256

  // Phase 1: 256 waves (one per sample)
  k_states<<<32, 256, 0, stream>>>(X, params, PsiR, PsiI);
  // Phase 2: 16384 waves (one per (j, s))
  k_unitary<<<2048, 256, 0, stream>>>(X, params, Ar, Aim);
  // Phase 3: one block per j, async-LDS staged WMMA complex GEMM
  k_gemm<<<256, 256, 0, stream>>>(Ar, Aim, PsiR, PsiI, Kraw);
  // Phase 4: symmetrize
  k_symmetrize<<<256, 256, 0, stream>>>(Kraw, out);
}